// Rotate_world_2800318677676
// MI455X (gfx1250) — compile-verified
//
#include <hip/hip_runtime.h>
#include <hip/hip_bf16.h>

typedef __attribute__((ext_vector_type(2))) float v2f;
typedef __attribute__((ext_vector_type(8))) float v8f;

#define Gq 64
#define G3q (Gq*Gq*Gq)

// Element [a][c] of inv(P) for a rigid pose P = [R|t; 0 0 0 1].
// inv = [R^T | -R^T t; 0 0 0 1]
__device__ inline float rigid_inv_elem(const float* __restrict__ P, int a, int c) {
    if (a == 3) return (c == 3) ? 1.0f : 0.0f;
    if (c < 3)  return P[c * 4 + a];                 // (R^T)[a][c] = R[c][a]
    // c == 3: -(R^T t)[a]
    return -(P[0 * 4 + a] * P[3] + P[1 * 4 + a] * P[7] + P[2 * 4 + a] * P[11]);
}

// One wave32. Computes T_n = pose0_b @ inv(pose1_{b,k+1}) for n = b*4+k, n=0..7,
// using V_WMMA_F32_16X16X4_F32: A packs 4 pose0 matrices (16 rows x K=4),
// B packs 4 inv(pose1) matrices (K=4 x 16 cols); diagonal 4x4 blocks of D are
// the composed transforms. Two WMMA iterations cover all 8 transforms.
// Writes rows 0..2 (3x4 = 12 floats) per transform to Tout.
__global__ void pose_compose_wmma(const float* __restrict__ cam,
                                  float* __restrict__ Tout) {
    const int lane = threadIdx.x;        // 0..31
    const int half = lane >> 4;          // 0: K=0,1 ; 1: K=2,3
    const int l16  = lane & 15;
    const int kb   = 2 * half;

    for (int g = 0; g < 2; ++g) {
        const int im = 4 * g + (l16 >> 2);   // global transform id for this lane's slot
        const int b  = im >> 2;              // batch
        const int k  = im & 3;               // frame-1
        const float* P0 = cam + (size_t)(b * 5 + 0) * 16;
        const float* P1 = cam + (size_t)(b * 5 + k + 1) * 16;

        const int ra = l16 & 3;              // row within A's 4x4 block
        v2f A;
        A.x = P0[ra * 4 + kb];
        A.y = P0[ra * 4 + kb + 1];

        const int cb = l16 & 3;              // col within B's 4x4 block
        v2f Bv;
        Bv.x = rigid_inv_elem(P1, kb,     cb);
        Bv.y = rigid_inv_elem(P1, kb + 1, cb);

        v8f Cz = {};
        v8f D = __builtin_amdgcn_wmma_f32_16x16x4_f32(
            /*neg_a=*/false, A, /*neg_b=*/false, Bv,
            /*c_mod=*/(short)0, Cz, /*reuse_a=*/false, /*reuse_b=*/false);

        // Lane holds D[vgpr + 8*half][l16].  For the diagonal block containing
        // column n = l16 (local matrix il = n>>2), rows 4*il + r live at
        // d-index (il&1)*4 + r for the lanes that hold them:
        //   half==0 -> il 0,1 (n<8);  half==1 -> il 2,3 (n>=8).
        const int n  = l16;
        const int il = n >> 2;
        const int c  = n & 3;
        const bool contrib = (half == 0) ? (n < 8) : (n >= 8);
        const bool hi = (il & 1) != 0;
        const float e0 = hi ? D[4] : D[0];
        const float e1 = hi ? D[5] : D[1];
        const float e2 = hi ? D[6] : D[2];
        if (contrib) {
            const int gi = 4 * g + il;
            Tout[gi * 12 + 0 * 4 + c] = e0;
            Tout[gi * 12 + 1 * 4 + c] = e1;
            Tout[gi * 12 + 2 * 4 + c] = e2;
        }
    }
}

// One thread per output point; loops over the 16 channels.
// 1024 blocks per transform n, so n (and the T row / slice base) is
// block-uniform -> scalar loads and SGPR-base addressing.
__global__ void warp_kernel(const float* __restrict__ vox,
                            const float* __restrict__ T,
                            float* __restrict__ out) {
    const int n = blockIdx.x >> 10;                          // 0..7 (uniform)
    const int p = ((blockIdx.x & 1023) << 8) | threadIdx.x;  // 0..G3-1
    const int x =  p        & 63;
    const int y = (p >> 6)  & 63;
    const int z =  p >> 12;

    const float* Tn = T + n * 12;                            // scalar loads
    const float lx = ((float)x - 31.5f) * (1.0f / 64.0f);
    const float ly = ((float)y - 31.5f) * (1.0f / 64.0f);
    const float lz = ((float)z - 31.5f) * (1.0f / 64.0f);

    const float cx = Tn[0] * lx + Tn[1] * ly + Tn[2]  * lz + Tn[3];
    const float cy = Tn[4] * lx + Tn[5] * ly + Tn[6]  * lz + Tn[7];
    const float cz = Tn[8] * lx + Tn[9] * ly + Tn[10] * lz + Tn[11];

    const float inv_m = 64.0f / 31.5f;       // 1/m, m = 31.5/64
    // ix = ((g+1)*W - 1) * 0.5 with W=H=D=64
    const float ix = (cx * inv_m + 1.0f) * 32.0f - 0.5f;
    const float iy = (cy * inv_m + 1.0f) * 32.0f - 0.5f;
    const float iz = (cz * inv_m + 1.0f) * 32.0f - 0.5f;

    const float fx = floorf(ix), fy = floorf(iy), fz = floorf(iz);
    const float tx = ix - fx,  ty = iy - fy,  tz = iz - fz;
    const int x0 = (int)fx, y0 = (int)fy, z0 = (int)fz;
    const int x1 = x0 + 1,  y1 = y0 + 1,  z1 = z0 + 1;

    // per-axis weights, zeroed when the corner is out of bounds
    const float wx0 = (1.0f - tx) * ((x0 >= 0 && x0 < 64) ? 1.0f : 0.0f);
    const float wx1 =         tx  * ((x1 >= 0 && x1 < 64) ? 1.0f : 0.0f);
    const float wy0 = (1.0f - ty) * ((y0 >= 0 && y0 < 64) ? 1.0f : 0.0f);
    const float wy1 =         ty  * ((y1 >= 0 && y1 < 64) ? 1.0f : 0.0f);
    const float wz0 = (1.0f - tz) * ((z0 >= 0 && z0 < 64) ? 1.0f : 0.0f);
    const float wz1 =         tz  * ((z1 >= 0 && z1 < 64) ? 1.0f : 0.0f);

    const int cx0 = x0 < 0 ? 0 : (x0 > 63 ? 63 : x0);
    const int cx1 = x1 < 0 ? 0 : (x1 > 63 ? 63 : x1);
    const int cy0 = y0 < 0 ? 0 : (y0 > 63 ? 63 : y0);
    const int cy1 = y1 < 0 ? 0 : (y1 > 63 ? 63 : y1);
    const int cz0 = z0 < 0 ? 0 : (z0 > 63 ? 63 : z0);
    const int cz1 = z1 < 0 ? 0 : (z1 > 63 ? 63 : z1);

    float w[8];
    int   off[8];
    w[0] = wz0 * wy0 * wx0;  off[0] = (cz0 << 12) | (cy0 << 6) | cx0;
    w[1] = wz0 * wy0 * wx1;  off[1] = (cz0 << 12) | (cy0 << 6) | cx1;
    w[2] = wz0 * wy1 * wx0;  off[2] = (cz0 << 12) | (cy1 << 6) | cx0;
    w[3] = wz0 * wy1 * wx1;  off[3] = (cz0 << 12) | (cy1 << 6) | cx1;
    w[4] = wz1 * wy0 * wx0;  off[4] = (cz1 << 12) | (cy0 << 6) | cx0;
    w[5] = wz1 * wy0 * wx1;  off[5] = (cz1 << 12) | (cy0 << 6) | cx1;
    w[6] = wz1 * wy1 * wx0;  off[6] = (cz1 << 12) | (cy1 << 6) | cx0;
    w[7] = wz1 * wy1 * wx1;  off[7] = (cz1 << 12) | (cy1 << 6) | cx1;

    const int b = n >> 2, k = n & 3;                         // uniform
    const size_t slice = (size_t)(b * 5 + k + 1) * 16 * G3q; // uniform (SGPR base)
    const float* __restrict__ src = vox + slice;
    float* __restrict__ dst = out + slice + p;

    for (int c = 0; c < 16; ++c) {
        const float* __restrict__ sc = src + (size_t)c * G3q;
        if (c < 15) {
            // Next channel is exactly G3 floats (1 MB) away: prefetch its two z-rows.
            __builtin_prefetch(sc + G3q + off[0], 0, 3);
            __builtin_prefetch(sc + G3q + off[4], 0, 3);
        }
        float acc = 0.0f;
#pragma unroll
        for (int j = 0; j < 8; ++j) acc += w[j] * sc[off[j]];
        dst[(size_t)c * G3q] = acc;
    }
}

// Pass-through of frame 0 for both batches: B128 streaming copy.
// 4096 blocks per batch -> b is block-uniform (SGPR base addressing).
__global__ void copy_first(const float4* __restrict__ vox, float4* __restrict__ out) {
    const int b = blockIdx.x >> 12;                               // 0..1 (uniform)
    const int r = ((blockIdx.x & 4095) << 8) | threadIdx.x;       // 0..2^20-1
    const size_t o = (size_t)b * (5 * 16 * (G3q / 4)) + r;        // frame-0 offset
    out[o] = vox[o];
}

extern "C" void kernel_launch(void* const* d_in, const int* in_sizes, int n_in,
                              void* d_out, int out_size, void* d_ws, size_t ws_size,
                              hipStream_t stream) {
    const float* vox = (const float*)d_in[0];   // (2,5,16,64,64,64) f32
    const float* cam = (const float*)d_in[1];   // (2,5,4,4) f32
    float* out = (float*)d_out;                 // (2,5,16,64,64,64) f32
    float* Tws = (float*)d_ws;                  // 8 * 12 floats

    (void)in_sizes; (void)n_in; (void)out_size; (void)ws_size;

    pose_compose_wmma<<<1, 32, 0, stream>>>(cam, Tws);

    warp_kernel<<<8 * (G3q / 256), 256, 0, stream>>>(vox, Tws, out);   // 8192 blocks

    copy_first<<<2 * 16 * (G3q / 4) / 256, 256, 0, stream>>>(          // 8192 blocks
        (const float4*)vox, (float4*)out);
}